// PathFeatureExtractor_51848845197659
// MI455X (gfx1250) — compile-verified
//
#include <hip/hip_runtime.h>

#define EPSF      1e-10f
#define XI0F      (0.235f * 0.235f)
#define SQRT_XI0F 0.235f
#define ETA2F     (1.9f * 1.9f)
#define NPATH     2048
#define DTF       (1.0f / 2048.0f)
#define INFF      __builtin_inff()

typedef unsigned int v4u __attribute__((ext_vector_type(4)));
typedef int          v4i __attribute__((ext_vector_type(4)));
typedef int          v8i __attribute__((ext_vector_type(8)));
typedef float        v4f __attribute__((ext_vector_type(4)));   // clang-native float4

// One block per batch row. 256 threads = 8 waves (wave32); each thread owns 8
// consecutive timesteps. S row (2049 f32) is DMA'd into LDS by the Tensor Data
// Mover; all scans run wave32-shuffle + LDS combine; windowed features are LDS
// lookups against the resident prefix-sum row. Streaming output uses TH=NT
// stores to keep 402 MB of write-once data out of WGP$/L2.
__global__ __launch_bounds__(256)
void path_feat_kernel(const float* __restrict__ S,
                      const float* __restrict__ dprev,
                      float* __restrict__ out)
{
    __shared__ float lds_l[NPATH + 1];    // S row, then log(clip(S)) in place
    __shared__ float lds_cum[NPATH + 1];  // cum_pad: prefix of squared log-returns
    __shared__ float lds_lrv[NPATH];      // log(clip(rv_short))
    __shared__ float wsum[8], wmaxs[8], wmins[8];

    const int t    = threadIdx.x;
    const int lane = t & 31;
    const int wid  = t >> 5;
    const int b    = blockIdx.x;
    const int k0   = t * 8;
    const float* Srow = S + (size_t)b * (NPATH + 1);

    // ---- Phase 1: S row -> LDS via TDM async tensor load --------------------
#if __has_builtin(__builtin_amdgcn_tensor_load_to_lds)
    if (t < 32) {  // one wave issues the TDM op (EXEC-independent, wave-scoped)
        unsigned long long ga = (unsigned long long)(const void*)Srow;
        unsigned lds_off = (unsigned)(size_t)(void*)&lds_l[0];  // low 32b = LDS byte offset
        v4u g0;
        g0.x = 1u;                                    // count=1, user descriptor
        g0.y = lds_off;                               // lds_addr (bytes)
        g0.z = (unsigned)(ga & 0xFFFFFFFFu);          // global_addr[31:0]
        g0.w = (unsigned)((ga >> 32) & 0x1FFFFFFu)    // global_addr[56:32]
             | (2u << 30);                            // type = 2 ("image")
        v8i g1;
        g1[0] = (2 << 16);                            // data_size = 2 -> 4 bytes
        g1[1] = (int)(((NPATH + 1) & 0xFFFF) << 16);  // tensor_dim0[15:0]
        g1[2] = (int)(((NPATH + 1) >> 16) | (1u << 16)); // dim0[31:16] | tensor_dim1=1
        g1[3] = (int)(((NPATH + 1) & 0xFFFF) << 16);  // dim1[31:16]=0 | tile_dim0=2049
        g1[4] = 1;                                    // tile_dim1=1, tile_dim2=0
        g1[5] = (NPATH + 1);                          // tensor_dim0_stride lo32
        g1[6] = 0;
        g1[7] = 0;
        v4i gz = {0, 0, 0, 0};
#if defined(__clang_major__) && __clang_major__ >= 23
        v8i gz8 = {0, 0, 0, 0, 0, 0, 0, 0};
        __builtin_amdgcn_tensor_load_to_lds(g0, g1, gz, gz, gz8, 0);
#else
        __builtin_amdgcn_tensor_load_to_lds(g0, g1, gz, gz, 0);
#endif
        __builtin_amdgcn_s_wait_tensorcnt(0);
    }
    __syncthreads();
#else
    for (int i = t; i < NPATH + 1; i += 256) lds_l[i] = Srow[i];
    __syncthreads();
#endif

    // ---- Phase 2: log(clip(S)) in place (each thread touches only its slots)
    #pragma unroll
    for (int j = 0; j < 8; ++j) {
        float s = lds_l[k0 + j];
        lds_l[k0 + j] = logf(fmaxf(s, EPSF));
    }
    if (t == 255) {
        float s = lds_l[NPATH];
        lds_l[NPATH] = logf(fmaxf(s, EPSF));
    }
    __syncthreads();

    float lv[9];
    #pragma unroll
    for (int j = 0; j < 9; ++j) lv[j] = lds_l[k0 + j];
    const float l0 = lds_l[0];

    // ---- Phase 3: fused sum/max/min scans -----------------------------------
    float psum[8], pmax[8], pmin[8];
    float s = 0.f, mx = -INFF, mn = INFF;
    #pragma unroll
    for (int j = 0; j < 8; ++j) {
        float r = lv[j + 1] - lv[j];     // log return ending at k0+j+1
        s += r * r;        psum[j] = s;
        mx = fmaxf(mx, lv[j]); pmax[j] = mx;
        mn = fminf(mn, lv[j]); pmin[j] = mn;
    }
    const float tot_s = s;
    float ws = s, wmx = mx, wmn = mn;    // wave32 inclusive scans
    #pragma unroll
    for (int d = 1; d < 32; d <<= 1) {
        float as = __shfl_up(ws,  d, 32);
        float am = __shfl_up(wmx, d, 32);
        float an = __shfl_up(wmn, d, 32);
        if (lane >= d) { ws += as; wmx = fmaxf(wmx, am); wmn = fminf(wmn, an); }
    }
    if (lane == 31) { wsum[wid] = ws; wmaxs[wid] = wmx; wmins[wid] = wmn; }
    float ex_ws  = ws - tot_s;                       // wave-exclusive sum
    float ex_wmx = __shfl_up(wmx, 1, 32);
    float ex_wmn = __shfl_up(wmn, 1, 32);
    if (lane == 0) { ex_wmx = -INFF; ex_wmn = INFF; }
    __syncthreads();

    float base_s = 0.f, base_mx = -INFF, base_mn = INFF;
    for (int w = 0; w < wid; ++w) {
        base_s += wsum[w];
        base_mx = fmaxf(base_mx, wmaxs[w]);
        base_mn = fminf(base_mn, wmins[w]);
    }
    const float pre_s  = base_s + ex_ws;
    const float pre_mx = fmaxf(base_mx, ex_wmx);
    const float pre_mn = fminf(base_mn, ex_wmn);

    #pragma unroll
    for (int j = 0; j < 8; ++j) lds_cum[k0 + 1 + j] = pre_s + psum[j];
    if (t == 0) lds_cum[0] = 0.f;

    float rmaxv[8], rminv[8];
    #pragma unroll
    for (int j = 0; j < 8; ++j) {
        rmaxv[j] = fmaxf(pre_mx, pmax[j]) - l0;   // cummax(f2)
        rminv[j] = fminf(pre_mn, pmin[j]) - l0;   // cummin(f2)
    }
    __syncthreads();

    // ---- Phase 4: rv_short (w=5) and its log into LDS -----------------------
    float rv5[8];
    #pragma unroll
    for (int j = 0; j < 8; ++j) {
        int k = k0 + j;
        float v;
        if (k == 0) v = SQRT_XI0F;
        else {
            int st = (k > 5) ? (k - 5) : 0;
            float win = (float)(k - st);
            float sm  = lds_cum[k] - lds_cum[st];
            v = sqrtf(fmaxf(sm / (win * DTF + EPSF), EPSF));
        }
        rv5[j] = v;
        lds_lrv[k] = logf(fmaxf(v, EPSF));
    }
    __syncthreads();

    // ---- Phase 5: assemble 12 features, contiguous NT v4f stores ------------
    const float* dp = dprev + (size_t)b * NPATH + k0;
    v4f d0 = __builtin_nontemporal_load((const v4f*)dp);
    v4f d1 = __builtin_nontemporal_load(((const v4f*)dp) + 1);
    float dpr[8] = {d0.x, d0.y, d0.z, d0.w, d1.x, d1.y, d1.z, d1.w};

    float* orow = out + ((size_t)b * NPATH + k0) * 12;
    const float inv = 1.0f / SQRT_XI0F;
    #pragma unroll
    for (int j = 0; j < 8; ++j) {
        int k = k0 + j;
        float f1 = (float)k * DTF;
        float f2 = lv[j] - l0;
        float f3 = 1.0f - f1;
        float cumk = lds_cum[k];
        float sig5 = rv5[j], sig15, sig50;
        if (k == 0) { sig15 = SQRT_XI0F; sig50 = SQRT_XI0F; }
        else {
            int st15 = (k > 15) ? (k - 15) : 0;
            sig15 = sqrtf(fmaxf((cumk - lds_cum[st15]) / ((float)(k - st15) * DTF + EPSF), EPSF));
            int st50 = (k > 50) ? (k - 50) : 0;
            sig50 = sqrtf(fmaxf((cumk - lds_cum[st50]) / ((float)(k - st50) * DTF + EPSF), EPSF));
        }
        float R = fminf(fmaxf(sig5 / (sig50 + EPSF), 0.2f), 5.0f);

        float Q = 0.f;
        if (k >= 2) {                       // vov window over log(rv_short)
            int end = k - 1;
            int stq = (k - 16 > 0) ? (k - 16) : 0;
            float s2 = 0.f;
            float prev = lds_lrv[stq];
            for (int m = stq; m < end; ++m) {
                float cur = lds_lrv[m + 1];
                float d = cur - prev;
                s2 += d * d;
                prev = cur;
            }
            float win = (float)(end - stq);  // >= 1 here
            Q = s2 / (win * DTF + EPSF) / ETA2F;
            Q = fminf(fmaxf(Q, 0.f), 10.f);
        }
        float tsafe = fmaxf(f1, DTF);
        float qvn = fminf(fmaxf(cumk / (XI0F * tsafe), 0.f), 10.f);

        v4f o0 = {f1, f2, f3, dpr[j]};
        v4f o1 = {sig5 * inv, sig15 * inv, sig50 * inv, R};
        v4f o2 = {Q, rmaxv[j], rminv[j], qvn};
        v4f* op = (v4f*)(orow + j * 12);
        __builtin_nontemporal_store(o0, op + 0);
        __builtin_nontemporal_store(o1, op + 1);
        __builtin_nontemporal_store(o2, op + 2);
    }
}

extern "C" void kernel_launch(void* const* d_in, const int* in_sizes, int n_in,
                              void* d_out, int out_size, void* d_ws, size_t ws_size,
                              hipStream_t stream) {
    const float* S     = (const float*)d_in[0];
    const float* dprev = (const float*)d_in[1];
    float* out = (float*)d_out;
    int batch = in_sizes[0] / (NPATH + 1);   // 4096
    path_feat_kernel<<<batch, 256, 0, stream>>>(S, dprev, out);
}